// DGCNN_ocardo_53996328846140
// MI455X (gfx1250) — compile-verified
//
#include <hip/hip_runtime.h>
#include <hip/hip_bf16.h>

// ---------------------------------------------------------------------------
// DGCNN on MI455X (gfx1250, wave32).
//
// EdgeConv refactor:  relu(cat([xi, xj-xi]) @ W + b)
//                   = relu( (xi @ (Wtop-Wbot) + b) + (xj @ Wbot) )
// => per layer: one node-level fp32 WMMA GEMM  H[N,64] @ Wc[64,128] -> {A,P},
//    then an L2-bound gather/max edge phase (dst is repeat(arange(N),16),
//    so segment_max = max over 16 consecutive edges).
// ---------------------------------------------------------------------------

typedef float v2f __attribute__((ext_vector_type(2)));
typedef float v8f __attribute__((ext_vector_type(8)));

#define N_NODES 100000
#define PB_LAYER 8192   // 16 ksteps * 2 halves * 128 cols * 2 floats

// ---------------------------------------------------------------------------
// Pre-pack weights into WMMA-B fragment order and zero the global-max slot.
// PB[s][h][n][d] = Wc[4s+2h+d][n], where for layers 2..5 (W is [128,64]):
//   Wc[k][n] = (n<64) ? W[k][n]-W[64+k][n]  (A half, gets +bias)
//            :          W[64+k][n-64]       (P half)
// and for the head (l==4): Wc[k][n] = Wa[k][n] (top 64 rows of Wa[128,128]).
// ---------------------------------------------------------------------------
__global__ __launch_bounds__(256) void prep_k(
    const float* __restrict__ W2, const float* __restrict__ W3,
    const float* __restrict__ W4, const float* __restrict__ W5,
    const float* __restrict__ Wa, float* __restrict__ PB,
    unsigned* __restrict__ g)
{
  int tid = blockIdx.x * 256 + threadIdx.x;
  if (tid < 64) g[tid] = 0u;                       // identity for max of relu(..) >= 0
  if (tid >= 5 * PB_LAYER) return;
  int l = tid / PB_LAYER, r = tid % PB_LAYER;
  int d = r & 1;
  int n = (r >> 1) & 127;
  int h = (r >> 8) & 1;
  int s = r >> 9;
  int k = 4 * s + 2 * h + d;                        // 0..63
  float v;
  if (l < 4) {
    const float* W = (l == 0) ? W2 : (l == 1) ? W3 : (l == 2) ? W4 : W5;
    v = (n < 64) ? (W[k * 64 + n] - W[(64 + k) * 64 + n])
                 : W[(64 + k) * 64 + (n - 64)];
  } else {
    v = Wa[k * 128 + n];                            // Wa_top
  }
  PB[tid] = v;
}

// ---------------------------------------------------------------------------
// Layer 1 (C=3): tiny 6x64 MLP done with VALU.  A = x@(W1top-W1bot)+b1,
// P = x@W1bot.  One thread per (node, channel).
// ---------------------------------------------------------------------------
__global__ __launch_bounds__(256) void layer1_ap(
    const float* __restrict__ x, const float* __restrict__ W1,
    const float* __restrict__ b1, float* __restrict__ A, float* __restrict__ P)
{
  int tid = blockIdx.x * 256 + threadIdx.x;         // N*64 threads
  int n = tid >> 6, c = tid & 63;
  float x0 = x[n * 3 + 0], x1 = x[n * 3 + 1], x2 = x[n * 3 + 2];
  float wt0 = W1[0 * 64 + c], wt1 = W1[1 * 64 + c], wt2 = W1[2 * 64 + c];
  float wb0 = W1[3 * 64 + c], wb1 = W1[4 * 64 + c], wb2 = W1[5 * 64 + c];
  float p = x0 * wb0 + x1 * wb1 + x2 * wb2;
  float a = b1[c] + x0 * (wt0 - wb0) + x1 * (wt1 - wb1) + x2 * (wt2 - wb2);
  A[tid] = a;
  P[tid] = p;
}

// ---------------------------------------------------------------------------
// fp32 WMMA GEMM: {A,P}[32x128 tile] = H[32x64] @ Wc[64x128] (+bias on A half)
// 8 waves/block: wave (wr=w>>2, wc=w&3) computes rows 16*wr, cols 32*wc..+31
// as two 16x16 tiles via 16 K-steps of v_wmma_f32_16x16x4_f32.
// A-fragment (16x4 f32): lane<16 holds {K=4s,4s+1} of row M=lane,
//                        lane>=16 holds {K=4s+2,4s+3} of row M=lane-16.
// B-fragment comes pre-packed (contiguous b64 per lane) from PB.
// ---------------------------------------------------------------------------
__global__ __launch_bounds__(256) void gemm_ap_k(
    const float* __restrict__ H, const float* __restrict__ PB,
    const float* __restrict__ bias, float* __restrict__ A, float* __restrict__ P)
{
  const int w = threadIdx.x >> 5;
  const int lane = threadIdx.x & 31;
  const int wr = w >> 2, wc = w & 3;
  const int n0 = blockIdx.x * 32 + wr * 16;
  const int lhalf = lane >> 4, l15 = lane & 15;
  const int cb0 = wc * 32, cb1 = wc * 32 + 16;

  v8f acc0 = {0.f, 0.f, 0.f, 0.f, 0.f, 0.f, 0.f, 0.f};
  v8f acc1 = {0.f, 0.f, 0.f, 0.f, 0.f, 0.f, 0.f, 0.f};

  const float* arow = H + (size_t)(n0 + l15) * 64 + 2 * lhalf;
#pragma unroll
  for (int s = 0; s < 16; ++s) {
    v2f af = *(const v2f*)(arow + 4 * s);
    v2f bf0 = *(const v2f*)(PB + (((s * 2 + lhalf) * 128) + cb0 + l15) * 2);
    v2f bf1 = *(const v2f*)(PB + (((s * 2 + lhalf) * 128) + cb1 + l15) * 2);
    acc0 = __builtin_amdgcn_wmma_f32_16x16x4_f32(false, af, false, bf0,
                                                 (short)0, acc0, false, false);
    acc1 = __builtin_amdgcn_wmma_f32_16x16x4_f32(false, af, false, bf1,
                                                 (short)0, acc1, false, false);
  }

  // D tile layout: VGPR r -> (M = r + 8*lhalf, N = cb + l15)
#pragma unroll
  for (int r = 0; r < 8; ++r) {
    int row = n0 + r + lhalf * 8;
    int c0 = cb0 + l15, c1 = cb1 + l15;
    float v0 = acc0[r], v1 = acc1[r];
    if (c0 < 64) A[(size_t)row * 64 + c0] = v0 + bias[c0];
    else         P[(size_t)row * 64 + (c0 - 64)] = v0;
    if (c1 < 64) A[(size_t)row * 64 + c1] = v1 + bias[c1];
    else         P[(size_t)row * 64 + (c1 - 64)] = v1;
  }
}

// ---------------------------------------------------------------------------
// Edge phase: h[i] = max_{e in 16 edges of i} relu(A[i] + P[src[e]]).
// One wave per node, lane owns channels {2*lane, 2*lane+1} (float2 = b64).
// Since relu>=0, init 0 is the exact identity and the final relu is folded in.
// ---------------------------------------------------------------------------
__global__ __launch_bounds__(256) void edge_max_k(
    const float* __restrict__ A, const float* __restrict__ P,
    const int* __restrict__ src, float* __restrict__ h)
{
  int node = blockIdx.x * 8 + (threadIdx.x >> 5);
  int lane = threadIdx.x & 31;
  float2 a = *(const float2*)(A + (size_t)node * 64 + lane * 2);
  float m0 = 0.0f, m1 = 0.0f;
  const int* se = src + (size_t)node * 16;
#pragma unroll
  for (int e = 0; e < 16; ++e) {
    int j = se[e];
    float2 p = *(const float2*)(P + (size_t)j * 64 + lane * 2);
    m0 = fmaxf(m0, a.x + p.x);
    m1 = fmaxf(m1, a.y + p.y);
  }
  float2 o; o.x = m0; o.y = m1;
  *(float2*)(h + (size_t)node * 64 + lane * 2) = o;
}

// ---------------------------------------------------------------------------
// Global 64-channel max over all nodes. h >= 0 (relu), so u32-bit atomicMax
// is exact. g pre-zeroed by prep_k every call (graph-replay safe).
// ---------------------------------------------------------------------------
__global__ __launch_bounds__(256) void global_max_k(
    const float* __restrict__ h, unsigned* __restrict__ g)
{
  __shared__ float red[256];
  int c = threadIdx.x & 63, grp = threadIdx.x >> 6;
  float m = 0.0f;
  for (int n = blockIdx.x * 4 + grp; n < N_NODES; n += gridDim.x * 4)
    m = fmaxf(m, h[(size_t)n * 64 + c]);
  red[threadIdx.x] = m;
  __syncthreads();
  if (threadIdx.x < 64) {
    m = fmaxf(fmaxf(red[threadIdx.x], red[threadIdx.x + 64]),
              fmaxf(red[threadIdx.x + 128], red[threadIdx.x + 192]));
    atomicMax(g + c, __float_as_uint(m));
  }
}

// gvec[c] = ba[c] + sum_k g[k] * Wa[64+k][c]   (node-invariant head bias)
__global__ void gvec_k(const unsigned* __restrict__ g,
                       const float* __restrict__ Wa,
                       const float* __restrict__ ba, float* __restrict__ gv)
{
  int c = threadIdx.x;  // 128 threads
  float s = ba[c];
  for (int k = 0; k < 64; ++k)
    s += __uint_as_float(g[k]) * Wa[(64 + k) * 128 + c];
  gv[c] = s;
}

// ---------------------------------------------------------------------------
// Head: t = relu(h @ Wa_top + gvec) [32x128 WMMA tile -> LDS], then per-node
// 128->3 matvec with Wb,bb and the residual  out = x + t@Wb + bb.
// ---------------------------------------------------------------------------
__global__ __launch_bounds__(256) void final_k(
    const float* __restrict__ h, const float* __restrict__ PBa,
    const float* __restrict__ gv, const float* __restrict__ Wb,
    const float* __restrict__ bb, const float* __restrict__ x,
    float* __restrict__ out)
{
  __shared__ float T[32][132];

  const int w = threadIdx.x >> 5;
  const int lane = threadIdx.x & 31;
  const int wr = w >> 2, wc = w & 3;
  const int n0 = blockIdx.x * 32 + wr * 16;
  const int lhalf = lane >> 4, l15 = lane & 15;
  const int cb0 = wc * 32, cb1 = wc * 32 + 16;

  v8f acc0 = {0.f, 0.f, 0.f, 0.f, 0.f, 0.f, 0.f, 0.f};
  v8f acc1 = {0.f, 0.f, 0.f, 0.f, 0.f, 0.f, 0.f, 0.f};

  const float* arow = h + (size_t)(n0 + l15) * 64 + 2 * lhalf;
#pragma unroll
  for (int s = 0; s < 16; ++s) {
    v2f af = *(const v2f*)(arow + 4 * s);
    v2f bf0 = *(const v2f*)(PBa + (((s * 2 + lhalf) * 128) + cb0 + l15) * 2);
    v2f bf1 = *(const v2f*)(PBa + (((s * 2 + lhalf) * 128) + cb1 + l15) * 2);
    acc0 = __builtin_amdgcn_wmma_f32_16x16x4_f32(false, af, false, bf0,
                                                 (short)0, acc0, false, false);
    acc1 = __builtin_amdgcn_wmma_f32_16x16x4_f32(false, af, false, bf1,
                                                 (short)0, acc1, false, false);
  }

#pragma unroll
  for (int r = 0; r < 8; ++r) {
    int lr = wr * 16 + r + lhalf * 8;
    int c0 = cb0 + l15, c1 = cb1 + l15;
    T[lr][c0] = fmaxf(0.0f, acc0[r] + gv[c0]);
    T[lr][c1] = fmaxf(0.0f, acc1[r] + gv[c1]);
  }
  __syncthreads();

  int t = threadIdx.x;
  if (t < 96) {
    int row = t / 3, j = t % 3;
    float s = bb[j];
#pragma unroll 8
    for (int c = 0; c < 128; ++c) s += T[row][c] * Wb[c * 3 + j];
    size_t n = (size_t)blockIdx.x * 32 + row;
    out[n * 3 + j] = x[n * 3 + j] + s;
  }
}

// ---------------------------------------------------------------------------
extern "C" void kernel_launch(void* const* d_in, const int* in_sizes, int n_in,
                              void* d_out, int out_size, void* d_ws, size_t ws_size,
                              hipStream_t stream) {
  (void)in_sizes; (void)n_in; (void)out_size; (void)ws_size;
  const float* x  = (const float*)d_in[0];
  const int*   ei = (const int*)d_in[1];     // [2, E]; row 0 = src
  const float* W1 = (const float*)d_in[2];
  const float* b1 = (const float*)d_in[3];
  const float* W2 = (const float*)d_in[4];
  const float* b2 = (const float*)d_in[5];
  const float* W3 = (const float*)d_in[6];
  const float* b3 = (const float*)d_in[7];
  const float* W4 = (const float*)d_in[8];
  const float* b4 = (const float*)d_in[9];
  const float* W5 = (const float*)d_in[10];
  const float* b5 = (const float*)d_in[11];
  const float* Wa = (const float*)d_in[12];
  const float* ba = (const float*)d_in[13];
  const float* Wb = (const float*)d_in[14];
  const float* bb = (const float*)d_in[15];
  float* out = (float*)d_out;

  const int N = N_NODES;
  float* ws = (float*)d_ws;
  float*    A  = ws;                          // N*64
  float*    Pp = A  + (size_t)N * 64;         // N*64
  float*    h0 = Pp + (size_t)N * 64;         // N*64
  float*    h1 = h0 + (size_t)N * 64;         // N*64
  float*    PB = h1 + (size_t)N * 64;         // 5 * 8192
  unsigned* g  = (unsigned*)(PB + 5 * PB_LAYER);  // 64
  float*    gv = (float*)(g + 64);            // 128

  const int* src = ei;  // edge_index[0]

  prep_k<<<(5 * PB_LAYER + 255) / 256, 256, 0, stream>>>(W2, W3, W4, W5, Wa, PB, g);
  layer1_ap<<<(N * 64) / 256, 256, 0, stream>>>(x, W1, b1, A, Pp);
  edge_max_k<<<N / 8, 256, 0, stream>>>(A, Pp, src, h0);

  const float* bs[4] = {b2, b3, b4, b5};
  float* hin = h0;
  float* hout = h1;
  for (int l = 0; l < 4; ++l) {
    gemm_ap_k<<<N / 32, 256, 0, stream>>>(hin, PB + l * PB_LAYER, bs[l], A, Pp);
    edge_max_k<<<N / 8, 256, 0, stream>>>(A, Pp, src, hout);
    float* tmp = hin; hin = hout; hout = tmp;
  }
  // hin == h after 5 EdgeConv layers

  global_max_k<<<256, 256, 0, stream>>>(hin, g);
  gvec_k<<<1, 128, 0, stream>>>(g, Wa, ba, gv);
  final_k<<<N / 32, 256, 0, stream>>>(hin, PB + 4 * PB_LAYER, gv, Wb, bb, x, out);
}